// SolverInTheLoop_62895501083203
// MI455X (gfx1250) — compile-verified
//
#include <hip/hip_runtime.h>
#include <stdint.h>

// ---------------- constants (match reference, computed in double, cast to f32) ----
constexpr float SIGMA3 = (float)(3.0 / 359.0 / 3.14159265358979323846); // quintic norm
constexpr float ETA_IJ = (float)(2.0 * 0.01 * 0.01 / (2.0 * 0.01 + 1e-8)); // scalar eta
constexpr float EPSF   = 1e-8f;

#define TPB   256
#define TILE  2048                  // edges per staged tile
#define EPT   (TILE / TPB)          // 8 edges per thread
#define EDGE_GRID 1024              // persistent blocks for edge kernels

__device__ __forceinline__ float pow5f(float x) { float x2 = x * x; return (x2 * x2) * x; }
__device__ __forceinline__ float pow4f(float x) { float x2 = x * x; return x2 * x2; }

// Hardware no-return f32 atomic add (VGLOBAL GLOBAL_ATOMIC_ADD_F32, no CAS loop).
__device__ __forceinline__ void hw_atomic_add_f32(float* p, float v) {
  uint64_t a = (uint64_t)(uintptr_t)p;
  asm volatile("global_atomic_add_f32 %0, %1, off" :: "v"(a), "v"(v));
}

// ---------------- async-to-LDS tile staging helper ------------------------------
// Stage TILE ints of i_s and TILE ints of j_s into LDS buffer b.
// Each thread issues 4 x global_load_async_to_lds_b128 (16B per lane, ASYNCcnt).
__device__ __forceinline__ void issue_tile_async(const int* i_s, const int* j_s,
                                                 int tile, uint32_t lds_base, int b,
                                                 int tid) {
  uint64_t gi = (uint64_t)(uintptr_t)(i_s + (size_t)tile * TILE);
  uint64_t gj = (uint64_t)(uintptr_t)(j_s + (size_t)tile * TILE);
  uint32_t lb = lds_base + (uint32_t)b * (2u * TILE * 4u);
#pragma unroll
  for (int s = 0; s < 2; ++s) {
    uint32_t c16 = (uint32_t)(tid + s * TPB) * 16u;
    uint32_t li = lb + c16;
    uint32_t lj = lb + TILE * 4u + c16;
    uint64_t ai = gi + c16;
    uint64_t aj = gj + c16;
    asm volatile("global_load_async_to_lds_b128 %0, %1, off"
                 :: "v"(li), "v"(ai) : "memory");
    asm volatile("global_load_async_to_lds_b128 %0, %1, off"
                 :: "v"(lj), "v"(aj) : "memory");
  }
}

// ---------------- per-edge force math (pass 2) ----------------------------------
__device__ __forceinline__ void force_edge(int i, int j, const float4* __restrict__ p8,
                                           float* __restrict__ out) {
  float4 a0 = p8[2 * (size_t)i + 0];
  float4 a1 = p8[2 * (size_t)i + 1];
  float4 b0 = p8[2 * (size_t)j + 0];
  float4 b1 = p8[2 * (size_t)j + 1];
  float dx = a0.x - b0.x, dy = a0.y - b0.y, dz = a0.z - b0.z;
  float dist = sqrtf(dx * dx + dy * dy + dz * dz);
  float q1 = fmaxf(0.f, 1.f - dist);
  float q2 = fmaxf(0.f, 2.f - dist);
  float q3 = fmaxf(0.f, 3.f - dist);
  float gw = (SIGMA3 * -5.f) * (pow4f(q3) - 6.f * pow4f(q2) + 15.f * pow4f(q1));
  float rho_i = a1.z, p_i = a1.w, rho_j = b1.z, p_j = b1.w;
  float inv_i = 1.f / rho_i, inv_j = 1.f / rho_j;
  float wv = inv_i * inv_i + inv_j * inv_j;
  float c = wv * gw / (dist + EPSF);
  float p_ij = (rho_j * p_i + rho_i * p_j) / (rho_i + rho_j);
  // u == v in the reference -> A_dr term is exactly zero; p_bg = 0 -> dvdt = 0
  float ux = a0.w - b0.w;                          // vx_i - vx_j
  float uy = a1.x - b1.x;
  float uz = a1.y - b1.y;
  float fx = c * (-p_ij * dx + ETA_IJ * ux);
  float fy = c * (-p_ij * dy + ETA_IJ * uy);
  float fz = c * (-p_ij * dz + ETA_IJ * uz);
  hw_atomic_add_f32(&out[(size_t)i * 8 + 0], fx);
  hw_atomic_add_f32(&out[(size_t)i * 8 + 1], fy);
  hw_atomic_add_f32(&out[(size_t)i * 8 + 2], fz);
}

// ---------------- kernel: force scatter (pass 2) --------------------------------
__global__ void __launch_bounds__(TPB)
k_forces(const int* __restrict__ i_s, const int* __restrict__ j_s,
         const float4* __restrict__ p8, float* __restrict__ out, int E) {
  __shared__ int sbuf[2][2][TILE];                 // [buf][i/j][edge] = 32 KB
  const int tid = threadIdx.x;
  const int ntiles = (E + TILE - 1) / TILE;
  const uint32_t lds_base = (uint32_t)(uintptr_t)(&sbuf[0][0][0]);

  int cur = blockIdx.x;
  int buf = 0;
  if (cur < ntiles && (long long)(cur + 1) * TILE <= (long long)E)
    issue_tile_async(i_s, j_s, cur, lds_base, 0, tid);

  for (; cur < ntiles; cur += gridDim.x, buf ^= 1) {
    int  nxt   = cur + (int)gridDim.x;
    bool nfull = (nxt < ntiles) && ((long long)(nxt + 1) * TILE <= (long long)E);
    bool cfull = (long long)(cur + 1) * TILE <= (long long)E;
    __syncthreads();                               // all reads of buf^1 (tile cur-G) done
    if (nfull) issue_tile_async(i_s, j_s, nxt, lds_base, buf ^ 1, tid);

    if (cfull) {
      if (nfull) asm volatile("s_wait_asynccnt 4" ::: "memory");
      else       asm volatile("s_wait_asynccnt 0" ::: "memory");
      __syncthreads();                             // whole tile visible in LDS
#pragma unroll
      for (int k = 0; k < EPT; ++k) {
        int i = sbuf[buf][0][tid + k * TPB];
        int j = sbuf[buf][1][tid + k * TPB];
        force_edge(i, j, p8, out);
      }
    } else {                                       // partial tail tile: direct loads
      int base = cur * TILE;
#pragma unroll
      for (int k = 0; k < EPT; ++k) {
        int e = base + tid + k * TPB;
        if (e < E) force_edge(i_s[e], j_s[e], p8, out);
      }
    }
  }
}

// ---------------- per-edge density math (pass 1) --------------------------------
__device__ __forceinline__ void density_edge(int i, int j, const float4* __restrict__ p8,
                                             float* __restrict__ out) {
  float4 a = p8[2 * (size_t)i];
  float4 b = p8[2 * (size_t)j];
  float dx = a.x - b.x, dy = a.y - b.y, dz = a.z - b.z;
  float d = sqrtf(dx * dx + dy * dy + dz * dz);
  float q1 = fmaxf(0.f, 1.f - d);
  float q2 = fmaxf(0.f, 2.f - d);
  float q3 = fmaxf(0.f, 3.f - d);
  float w = SIGMA3 * (pow5f(q3) - 6.f * pow5f(q2) + 15.f * pow5f(q1));
  hw_atomic_add_f32(&out[(size_t)i * 8 + 6], w);
}

// ---------------- kernel: density scatter (pass 1) ------------------------------
__global__ void __launch_bounds__(TPB)
k_density(const int* __restrict__ i_s, const int* __restrict__ j_s,
          const float4* __restrict__ p8, float* __restrict__ out, int E) {
  __shared__ int sbuf[2][2][TILE];
  const int tid = threadIdx.x;
  const int ntiles = (E + TILE - 1) / TILE;
  const uint32_t lds_base = (uint32_t)(uintptr_t)(&sbuf[0][0][0]);

  int cur = blockIdx.x;
  int buf = 0;
  if (cur < ntiles && (long long)(cur + 1) * TILE <= (long long)E)
    issue_tile_async(i_s, j_s, cur, lds_base, 0, tid);

  for (; cur < ntiles; cur += gridDim.x, buf ^= 1) {
    int  nxt   = cur + (int)gridDim.x;
    bool nfull = (nxt < ntiles) && ((long long)(nxt + 1) * TILE <= (long long)E);
    bool cfull = (long long)(cur + 1) * TILE <= (long long)E;
    __syncthreads();
    if (nfull) issue_tile_async(i_s, j_s, nxt, lds_base, buf ^ 1, tid);

    if (cfull) {
      if (nfull) asm volatile("s_wait_asynccnt 4" ::: "memory");
      else       asm volatile("s_wait_asynccnt 0" ::: "memory");
      __syncthreads();
#pragma unroll
      for (int k = 0; k < EPT; ++k) {
        int i = sbuf[buf][0][tid + k * TPB];
        int j = sbuf[buf][1][tid + k * TPB];
        density_edge(i, j, p8, out);
      }
    } else {
      int base = cur * TILE;
#pragma unroll
      for (int k = 0; k < EPT; ++k) {
        int e = base + tid + k * TPB;
        if (e < E) density_edge(i_s[e], j_s[e], p8, out);
      }
    }
  }
}

// ---------------- kernel: zero the output --------------------------------------
__global__ void k_zero(float4* out4, int n4) {
  int t = blockIdx.x * blockDim.x + threadIdx.x;
  int stride = gridDim.x * blockDim.x;
  float4 z = make_float4(0.f, 0.f, 0.f, 0.f);
  for (int k = t; k < n4; k += stride) out4[k] = z;
}

// ---------------- kernel: pack particle table {x,y,z,vx | vy,vz,rho,p} ----------
__global__ void k_pack(const float* __restrict__ abs_pos,
                       const float* __restrict__ vel,
                       float4* __restrict__ p8, int n) {
  int t = blockIdx.x * blockDim.x + threadIdx.x;
  if (t >= n) return;
  const float* b = abs_pos + (size_t)t * 15;       // [N,3,5], last column
  float x = b[4], y = b[9], z = b[14];
  const float* v = vel + (size_t)t * 3;
  p8[2 * (size_t)t + 0] = make_float4(x, y, z, v[0]);
  p8[2 * (size_t)t + 1] = make_float4(v[1], v[2], 0.f, 0.f);
}

// ---------------- kernel: EoS + finish particle table ---------------------------
__global__ void k_eos(float* __restrict__ out, float4* __restrict__ p8, int n) {
  int t = blockIdx.x * blockDim.x + threadIdx.x;
  if (t >= n) return;
  float rho = out[(size_t)t * 8 + 6];
  float p = 100.f * (rho - 1.f);                   // P_REF*((rho/1)^1 - 1) + 0
  out[(size_t)t * 8 + 7] = p;
  float4 s = p8[2 * (size_t)t + 1];
  s.z = rho; s.w = p;
  p8[2 * (size_t)t + 1] = s;
}

// ---------------- host launch ---------------------------------------------------
extern "C" void kernel_launch(void* const* d_in, const int* in_sizes, int n_in,
                              void* d_out, int out_size, void* d_ws, size_t ws_size,
                              hipStream_t stream) {
  const float* abs_pos = (const float*)d_in[0];
  const float* vel     = (const float*)d_in[1];
  const int*   i_s     = (const int*)d_in[2];
  const int*   j_s     = (const int*)d_in[3];
  float* out = (float*)d_out;

  const int N = in_sizes[1] / 3;   // vel_hist is [N,3]
  const int E = in_sizes[2];
  float4* p8 = (float4*)d_ws;      // 2 x float4 per particle = 32B * N

  const int n4 = out_size / 4;     // out is N*8 floats
  k_zero<<<512, TPB, 0, stream>>>((float4*)d_out, n4);
  k_pack<<<(N + TPB - 1) / TPB, TPB, 0, stream>>>(abs_pos, vel, p8, N);

  const int ntiles = (E + TILE - 1) / TILE;
  const int grid_e = ntiles < EDGE_GRID ? ntiles : EDGE_GRID;
  k_density<<<grid_e, TPB, 0, stream>>>(i_s, j_s, p8, out, E);
  k_eos<<<(N + TPB - 1) / TPB, TPB, 0, stream>>>(out, p8, N);
  k_forces<<<grid_e, TPB, 0, stream>>>(i_s, j_s, p8, out, E);
}